// Noise_operator_50225347560024
// MI455X (gfx1250) — compile-verified
//
#include <hip/hip_runtime.h>

typedef __attribute__((ext_vector_type(2))) float v2f;
typedef __attribute__((ext_vector_type(4))) float v4f;
typedef __attribute__((ext_vector_type(8))) float v8f;

// ---------------------------------------------------------------------------
// Full wave32 reduction using V_WMMA_F32_16X16X4_F32 (CDNA5 WMMA path).
// Layouts per CDNA5 ISA:
//   A 16x4 f32 : lane L<16 -> {A[L][0], A[L][1]},  lane L>=16 -> {A[L-16][2], A[L-16][3]}
//   D 16x16 f32: lane L<16 -> VGPR r = D[r][L],    lane L>=16 -> VGPR r = D[r+8][L-16]
// Step 1: A = lane sums (K-slot 0), B = ones  => D[m][n] = s_m + s_{m+16}
// Step 2: each lane sums its 8 D vgprs        => low half holds sum(rows 0-7),
//                                                high half holds sum(rows 8-15)
// Step 3: repeat WMMA with ones-B             => every lane holds the wave total.
// EXEC must be all 1s at the call site (callers guarantee reconvergence).
// ---------------------------------------------------------------------------
__device__ __forceinline__ float wave_reduce_wmma(float s) {
  v2f a;    a[0] = s;    a[1] = 0.0f;
  v2f ones; ones[0] = 1.0f; ones[1] = 1.0f;
  v8f c = {};
  v8f d1 = __builtin_amdgcn_wmma_f32_16x16x4_f32(
      false, a, false, ones, (short)0, c, false, false);
  float t = ((d1[0] + d1[1]) + (d1[2] + d1[3])) +
            ((d1[4] + d1[5]) + (d1[6] + d1[7]));
  v2f a2; a2[0] = t; a2[1] = 0.0f;
  v8f d2 = __builtin_amdgcn_wmma_f32_16x16x4_f32(
      false, a2, false, ones, (short)0, c, false, false);
  return d2[0];
}

// log(I0(x)), branchless (Abramowitz & Stegun 9.8.1 / 9.8.2).
// Large-x branch folds -0.5*log(x)+log(p) into log(p*rsqrt(x)) -> 2 trans total.
__device__ __forceinline__ float log_i0(float x) {
  float ax = __builtin_fabsf(x);
  // |x| <= 3.75 : polynomial in t = (x/3.75)^2
  float t = ax * 0.26666668f;  // 1/3.75
  t = t * t;
  float ps = __builtin_fmaf(t, __builtin_fmaf(t, __builtin_fmaf(t,
             __builtin_fmaf(t, __builtin_fmaf(t, __builtin_fmaf(t,
             0.0045813f, 0.0360768f), 0.2659732f), 1.2067492f),
             3.0899424f), 3.5156229f), 1.0f);
  float ls = __logf(ps);
  // |x| > 3.75 : u = 3.75/|x|;  log(I0) = x + log(poly(u) * rsqrt(x))
  float u = 3.75f * __frcp_rn(ax);
  float pl = __builtin_fmaf(u, __builtin_fmaf(u, __builtin_fmaf(u,
             __builtin_fmaf(u, __builtin_fmaf(u, __builtin_fmaf(u,
             __builtin_fmaf(u, __builtin_fmaf(u,
             0.00392377f, -0.01647633f), 0.02635537f), -0.02057706f),
             0.00916281f), -0.00157565f), 0.00225319f), 0.01328592f),
             0.39894228f);
  float ll = ax + __logf(pl * __frsqrt_rn(ax));
  return ax <= 3.75f ? ls : ll;
}

__device__ __forceinline__ float rician_term(float e, float o,
                                             float inv_var, float half_iv) {
  // accumulate (term1 - log_bessel); final sum equals -sum(log_bessel - term1)
  float bess_arg = e * o * inv_var;
  return __builtin_fmaf(e * e, half_iv, -log_i0(bess_arg));
}

#define NBLOCKS 1024
#define NTHREADS 256
#define NWAVES (NTHREADS / 32)

__global__ __launch_bounds__(NTHREADS)
void rician_nll_partial(const v4f* __restrict__ e4,
                        const v4f* __restrict__ o4,
                        const float* __restrict__ std_noise,
                        float* __restrict__ partials,
                        int n4, int n) {
  const float sd = std_noise[0];
  const float inv_var = __frcp_rn(sd * sd);
  const float half_iv = 0.5f * inv_var;

  float acc = 0.0f;
  const int stride = gridDim.x * blockDim.x;
  for (int i = blockIdx.x * blockDim.x + threadIdx.x; i < n4; i += stride) {
    v4f e = __builtin_nontemporal_load(e4 + i);  // global_load_b128 th:NT
    v4f o = __builtin_nontemporal_load(o4 + i);
    acc += rician_term(e[0], o[0], inv_var, half_iv);
    acc += rician_term(e[1], o[1], inv_var, half_iv);
    acc += rician_term(e[2], o[2], inv_var, half_iv);
    acc += rician_term(e[3], o[3], inv_var, half_iv);
  }
  // scalar tail (N % 4), handled by block 0; reconverges before the WMMA
  if (blockIdx.x == 0) {
    const float* e1 = (const float*)e4;
    const float* o1 = (const float*)o4;
    for (int i = n4 * 4 + threadIdx.x; i < n; i += NTHREADS)
      acc += rician_term(e1[i], o1[i], inv_var, half_iv);
  }

  float wsum = wave_reduce_wmma(acc);  // every lane holds the wave total

  __shared__ float lds[NWAVES];
  if ((threadIdx.x & 31) == 0) lds[threadIdx.x >> 5] = wsum;
  __syncthreads();
  if (threadIdx.x == 0) {
    float b = 0.0f;
#pragma unroll
    for (int w = 0; w < NWAVES; ++w) b += lds[w];
    partials[blockIdx.x] = b;
  }
}

__global__ __launch_bounds__(NTHREADS)
void rician_nll_final(const float* __restrict__ partials,
                      float* __restrict__ out, int n) {
  float acc = 0.0f;
  for (int i = threadIdx.x; i < n; i += NTHREADS)  // n % NTHREADS == 0: uniform
    acc += partials[i];

  float wsum = wave_reduce_wmma(acc);

  __shared__ float lds[NWAVES];
  if ((threadIdx.x & 31) == 0) lds[threadIdx.x >> 5] = wsum;
  __syncthreads();
  if (threadIdx.x == 0) {
    float b = 0.0f;
#pragma unroll
    for (int w = 0; w < NWAVES; ++w) b += lds[w];
    out[0] = b;
  }
}

extern "C" void kernel_launch(void* const* d_in, const int* in_sizes, int n_in,
                              void* d_out, int out_size, void* d_ws, size_t ws_size,
                              hipStream_t stream) {
  const float* est = (const float*)d_in[0];
  const float* obs = (const float*)d_in[1];
  const float* sn  = (const float*)d_in[2];
  float* out = (float*)d_out;
  float* partials = (float*)d_ws;  // NBLOCKS floats = 4 KB scratch

  const int n  = in_sizes[0];   // 33,554,432
  const int n4 = n / 4;

  rician_nll_partial<<<NBLOCKS, NTHREADS, 0, stream>>>(
      (const v4f*)est, (const v4f*)obs, sn, partials, n4, n);
  rician_nll_final<<<1, NTHREADS, 0, stream>>>(partials, out, NBLOCKS);
}